// DetectionLoss_18545668784810
// MI455X (gfx1250) — compile-verified
//
#include <hip/hip_runtime.h>
#include <hip/hip_bf16.h>

// ---------------- problem constants (from reference) ----------------
#define BB    4
#define NN    33600
#define GG    100
#define NCL   80
#define TOPKK 10
#define CENTER_R2 6.25f      // 2.5^2
#define BIGC  1.0e10f
#define EPSC  1e-7f

#define NB   ((NN + 255) / 256)      // 132 blocks of 256 per image
#define NT16 (NN / 16)               // 2100 anchor tiles of 16
#define CBPI ((NT16 + 7) / 8)        // 263 cand-blocks per image (8 waves/block)

typedef __attribute__((ext_vector_type(2))) float v2f;
typedef __attribute__((ext_vector_type(8))) float v8f;

__device__ __forceinline__ float sigmoidf_(float x) { return 1.0f / (1.0f + __expf(-x)); }
// numerically stable softplus
__device__ __forceinline__ float softplusf_(float x) {
    return fmaxf(x, 0.0f) + log1pf(expf(-fabsf(x)));
}

// ---------------- K0: init keys + counters ----------------
__global__ void k_init(unsigned long long* __restrict__ keys,
                       int* __restrict__ ncand, int* __restrict__ npos) {
    int i = blockIdx.x * blockDim.x + threadIdx.x;
    if (i < BB * NN) keys[i] = ~0ull;
    if (i < BB) { ncand[i] = 0; npos[i] = 0; }
}

// ---------------- K1: per-anchor sp_sum = sum_c softplus(sigmoid(pc)) ----------------
__global__ void k_prep(const float* __restrict__ pred, float* __restrict__ sp_sum) {
    int b = blockIdx.x / NB;
    int n = (blockIdx.x % NB) * 256 + threadIdx.x;
    if (n >= NN) return;
    const float* p = pred + ((size_t)b * NN + n) * (NCL + 5);
    float acc = 0.0f;
    #pragma unroll 4
    for (int c = 0; c < NCL; ++c) {
        float s = sigmoidf_(p[4 + c]);
        acc += log1pf(expf(s));          // softplus of value in (0,1): stable directly
    }
    sp_sum[(size_t)b * NN + n] = acc;
}

// ---------------- K2: candidate mask via WMMA f32 16x16x4 ----------------
// d^2(a,g) = |a|^2 + |g|^2 - 2 a.g  ==  [ax,ay,|a|^2,1] . [-2gx,-2gy,1,|g|^2]
// A 16x4 f32 layout: lanes 0-15 hold K=0,1 (vgpr0,1) for M=lane; lanes 16-31 hold K=2,3.
// B 4x16 f32 layout (symmetric): lanes 0-15 rows K=0,1 for N=lane&15; lanes 16-31 rows K=2,3.
// C/D 16x16 f32: vgpr v -> M=v (lanes 0-15) / M=v+8 (lanes 16-31), N = lane&15.
__global__ void __launch_bounds__(256)
k_cand(const float* __restrict__ anchors,   // (N,2)
       const float* __restrict__ gt_boxes,  // (B,G,4)
       unsigned char* __restrict__ cand,    // (B,N)
       int* __restrict__ ncand) {
    int b    = blockIdx.x / CBPI;
    int rem  = blockIdx.x % CBPI;
    int wave = threadIdx.x >> 5;
    int lane = threadIdx.x & 31;
    int tile = rem * 8 + wave;
    if (tile >= NT16) return;                 // wave-uniform exit: EXEC stays full for WMMA

    int half = lane >> 4;
    int lp   = lane & 15;
    int abase = tile * 16;

    // Build A fragment
    float ax = anchors[(abase + lp) * 2 + 0];
    float ay = anchors[(abase + lp) * 2 + 1];
    v2f afrag;
    if (half == 0) { afrag.x = ax;               afrag.y = ay;   }
    else           { afrag.x = ax * ax + ay * ay; afrag.y = 1.0f; }

    const float* gtb = gt_boxes + (size_t)b * GG * 4;
    unsigned int candmask = 0u;

    #pragma unroll
    for (int t = 0; t < 7; ++t) {             // 7*16 = 112 >= G=100
        int gidx = t * 16 + lp;
        int gc   = gidx < GG ? gidx : GG - 1;
        float gx = gtb[gc * 4 + 0];
        float gy = gtb[gc * 4 + 1];
        if (gidx >= GG) { gx = 1.0e7f; gy = 1.0e7f; }   // padded GT -> huge d^2
        v2f bfrag;
        if (half == 0) { bfrag.x = -2.0f * gx; bfrag.y = -2.0f * gy;        }
        else           { bfrag.x = 1.0f;       bfrag.y = gx * gx + gy * gy; }

        v8f c = {};
        v8f d2 = __builtin_amdgcn_wmma_f32_16x16x4_f32(
            false, afrag, false, bfrag, (short)0, c, false, false);

        #pragma unroll
        for (int v = 0; v < 8; ++v) {
            unsigned long long bal = __ballot(d2[v] < CENTER_R2);
            unsigned int m = (unsigned int)bal;           // wave32: low 32 bits
            if (m & 0xFFFFu)  candmask |= (1u << v);      // anchor abase+v
            if (m >> 16)      candmask |= (1u << (v + 8)); // anchor abase+v+8
        }
    }

    if (lane < 16)
        cand[(size_t)b * NN + abase + lane] = (unsigned char)((candmask >> lane) & 1u);
    if (lane == 0)
        atomicAdd(&ncand[b], __popc(candmask & 0xFFFFu));  // int atomics: deterministic
}

// ---------------- K3: per-(image,gt) cost scan + dynamic top-k + scatter ----------------
__global__ void __launch_bounds__(256)
k_assign(const float* __restrict__ pred,
         const float* __restrict__ gt_boxes,
         const int*   __restrict__ gt_classes,
         const float* __restrict__ sp_sum,
         const unsigned char* __restrict__ cand,
         const int*   __restrict__ ncand,
         unsigned long long* __restrict__ keys) {
    __shared__ float s_cost[256 * TOPKK];
    __shared__ unsigned int s_idx[256 * TOPKK];
    __shared__ float s_red[256];
    __shared__ unsigned int s_redp[256];
    __shared__ int s_dynk;

    int b = blockIdx.x / GG;
    int g = blockIdx.x % GG;
    int tid = threadIdx.x;

    // GT data (uniform -> scalar loads)
    const float* gt = gt_boxes + ((size_t)b * GG + g) * 4;
    float gx = gt[0], gy = gt[1], gw = gt[2], gh = gt[3];
    float gx1 = gx - gw * 0.5f, gy1 = gy - gh * 0.5f;
    float gx2 = gx + gw * 0.5f, gy2 = gy + gh * 0.5f;
    float garea = gw * gh;
    int   tcls = gt_classes[b * GG + g];
    int   nc_raw = ncand[b];
    bool  allcand = (nc_raw == 0);

    float lc[TOPKK]; unsigned int li[TOPKK];
    #pragma unroll
    for (int i = 0; i < TOPKK; ++i) { lc[i] = INFINITY; li[i] = 0xFFFFFFFFu; }

    float iousum = 0.0f;
    for (int n = tid; n < NN; n += 256) {
        const float* p = pred + ((size_t)b * NN + n) * (NCL + 5);
        float px = p[0], py = p[1], pw = p[2], ph = p[3];
        float x11 = px - pw * 0.5f, y11 = py - ph * 0.5f;
        float x12 = px + pw * 0.5f, y12 = py + ph * 0.5f;
        float iw = fmaxf(fminf(x12, gx2) - fmaxf(x11, gx1), 0.0f);
        float ih = fmaxf(fminf(y12, gy2) - fmaxf(y11, gy1), 0.0f);
        float inter = iw * ih;
        float iou = inter / (pw * ph + garea - inter + EPSC);
        bool cn = allcand || (cand[(size_t)b * NN + n] != 0);
        float iou_eff = cn ? iou : 0.0f;
        iousum += iou_eff;
        float cost;
        if (cn) {
            float score = sigmoidf_(p[4 + tcls]);
            cost = -logf(iou_eff + 1e-8f) + 3.0f * (sp_sum[(size_t)b * NN + n] - score);
        } else {
            cost = BIGC;
        }
        // register top-10 insertion (ascending cost, ties -> lower index)
        unsigned int un = (unsigned int)n;
        if (cost < lc[TOPKK - 1] || (cost == lc[TOPKK - 1] && un < li[TOPKK - 1])) {
            int j = TOPKK - 1;
            while (j > 0 && (cost < lc[j - 1] || (cost == lc[j - 1] && un < li[j - 1]))) {
                lc[j] = lc[j - 1]; li[j] = li[j - 1]; --j;
            }
            lc[j] = cost; li[j] = un;
        }
    }

    // ---- block reduce iou column sum -> dyn_k ----
    s_red[tid] = iousum;
    __syncthreads();
    for (int s = 128; s > 0; s >>= 1) {
        if (tid < s) s_red[tid] += s_red[tid + s];
        __syncthreads();
    }
    if (tid == 0) {
        int nc_eff = allcand ? NN : nc_raw;
        int kmax = min(TOPKK, nc_eff);
        int dk = (int)floorf(s_red[0]);
        dk = max(dk, 1);
        dk = min(dk, kmax);
        s_dynk = dk;
    }
    __syncthreads();

    // ---- merge 256 local top-10 lists ----
    #pragma unroll
    for (int i = 0; i < TOPKK; ++i) {
        s_cost[tid * TOPKK + i] = lc[i];
        s_idx [tid * TOPKK + i] = li[i];
    }
    __syncthreads();

    float tkc[TOPKK]; unsigned int tki[TOPKK];   // only thread 0's copy is used
    for (int r = 0; r < TOPKK; ++r) {
        float bc = INFINITY; unsigned int bp = (unsigned int)(tid * TOPKK);
        #pragma unroll
        for (int i = 0; i < TOPKK; ++i) {
            unsigned int ppos = (unsigned int)(tid * TOPKK + i);
            float c = s_cost[ppos];
            if (c < bc || (c == bc && s_idx[ppos] < s_idx[bp])) { bc = c; bp = ppos; }
        }
        s_red[tid] = bc; s_redp[tid] = bp;
        __syncthreads();
        for (int s = 128; s > 0; s >>= 1) {
            if (tid < s) {
                float c1 = s_red[tid], c2 = s_red[tid + s];
                unsigned int p1 = s_redp[tid], p2 = s_redp[tid + s];
                if (c2 < c1 || (c2 == c1 && s_idx[p2] < s_idx[p1])) {
                    s_red[tid] = c2; s_redp[tid] = p2;
                }
            }
            __syncthreads();
        }
        if (tid == 0) {
            unsigned int pwin = s_redp[0];
            tkc[r] = s_red[0];
            tki[r] = s_idx[pwin];
            s_cost[pwin] = INFINITY;
        }
        __syncthreads();
    }

    // ---- scatter: packed (cost_bits, g) key; atomicMin == reference dedup ----
    if (tid == 0) {
        int dk = s_dynk;
        for (int r = 0; r < dk; ++r) {
            unsigned long long key =
                (((unsigned long long)__float_as_uint(tkc[r])) << 32) |
                (unsigned long long)(unsigned int)g;
            atomicMin(&keys[(size_t)b * NN + tki[r]], key);
        }
    }
}

// ---------------- K4: per-anchor losses -> per-block partials ----------------
__global__ void __launch_bounds__(256)
k_losses(const float* __restrict__ pred,
         const float* __restrict__ gt_boxes,
         const int*   __restrict__ gt_classes,
         const unsigned long long* __restrict__ keys,
         float* __restrict__ box_part, float* __restrict__ cls_part,
         float* __restrict__ obj_part, int* __restrict__ npos) {
    __shared__ float sb[256], sc[256], so[256];
    __shared__ int   sn[256];
    int b = blockIdx.x / NB;
    int n = (blockIdx.x % NB) * 256 + threadIdx.x;
    int tid = threadIdx.x;

    float boxv = 0.0f, clsv = 0.0f, objv = 0.0f; int fgc = 0;
    if (n < NN) {
        const float* p = pred + ((size_t)b * NN + n) * (NCL + 5);
        float po = p[NCL + 4];
        unsigned long long key = keys[(size_t)b * NN + n];
        bool fg = (key != ~0ull);
        objv = softplusf_(po) - (fg ? po : 0.0f);
        if (fg) {
            fgc = 1;
            unsigned int g = (unsigned int)(key & 0xFFFFFFFFull);
            const float* gt = gt_boxes + ((size_t)b * GG + g) * 4;
            float gx = gt[0], gy = gt[1], w2 = gt[2], h2 = gt[3];
            float px = p[0], py = p[1], w1 = p[2], h1 = p[3];
            float x11 = px - w1 * 0.5f, y11 = py - h1 * 0.5f;
            float x12 = px + w1 * 0.5f, y12 = py + h1 * 0.5f;
            float x21 = gx - w2 * 0.5f, y21 = gy - h2 * 0.5f;
            float x22 = gx + w2 * 0.5f, y22 = gy + h2 * 0.5f;
            float iw = fmaxf(fminf(x12, x22) - fmaxf(x11, x21), 0.0f);
            float ih = fmaxf(fminf(y12, y22) - fmaxf(y11, y21), 0.0f);
            float inter = iw * ih;
            float uni = w1 * h1 + w2 * h2 - inter + EPSC;
            float iou = inter / uni;
            float cw = fmaxf(x12, x22) - fminf(x11, x21);
            float ch = fmaxf(y12, y22) - fminf(y11, y21);
            float c2 = cw * cw + ch * ch + EPSC;
            float dx = x11 + x12 - x21 - x22;
            float dy = y11 + y12 - y21 - y22;
            float rho2 = (dx * dx + dy * dy) * 0.25f;
            float dv = atanf(w2 / (h2 + EPSC)) - atanf(w1 / (h1 + EPSC));
            float v = (4.0f / 9.8696044010893586f) * dv * dv;
            float alpha = v / (v - iou + (1.0f + EPSC));
            float ciou = iou - (rho2 / c2 + v * alpha);
            boxv = 1.0f - ciou;

            int t = gt_classes[b * GG + (int)g];
            #pragma unroll 4
            for (int c = 0; c < NCL; ++c) {
                float x = p[4 + c];
                float tgt = (c == t) ? 1.0f : 0.0f;
                float bce = softplusf_(x) - x * tgt;
                float pp = sigmoidf_(x);
                float p_t = pp * tgt + (1.0f - pp) * (1.0f - tgt);
                float om = 1.0f - p_t;
                clsv += 0.25f * om * om * bce;
            }
        }
    }
    sb[tid] = boxv; sc[tid] = clsv; so[tid] = objv; sn[tid] = fgc;
    __syncthreads();
    for (int s = 128; s > 0; s >>= 1) {
        if (tid < s) { sb[tid] += sb[tid+s]; sc[tid] += sc[tid+s]; so[tid] += so[tid+s]; sn[tid] += sn[tid+s]; }
        __syncthreads();
    }
    if (tid == 0) {
        box_part[blockIdx.x] = sb[0];
        cls_part[blockIdx.x] = sc[0];
        obj_part[blockIdx.x] = so[0];
        atomicAdd(&npos[b], sn[0]);      // integer: deterministic
    }
}

// ---------------- K5: fixed-order final reduction ----------------
__global__ void k_final(const float* __restrict__ box_part,
                        const float* __restrict__ cls_part,
                        const float* __restrict__ obj_part,
                        const int*   __restrict__ npos,
                        float* __restrict__ out) {
    if (threadIdx.x != 0 || blockIdx.x != 0) return;
    float box = 0.0f, cls = 0.0f, obj = 0.0f;
    for (int i = 0; i < BB * NB; ++i) {
        box += box_part[i]; cls += cls_part[i]; obj += obj_part[i];
    }
    int np = npos[0] + npos[1] + npos[2] + npos[3];
    float n_pos = fmaxf((float)np, 1.0f);
    out[0] = 7.5f * box / n_pos + 0.5f * cls / n_pos + obj / (float)NN;
}

// ---------------- host launcher ----------------
extern "C" void kernel_launch(void* const* d_in, const int* in_sizes, int n_in,
                              void* d_out, int out_size, void* d_ws, size_t ws_size,
                              hipStream_t stream) {
    const float* pred       = (const float*)d_in[0];   // (B,N,85)
    const float* gt_boxes   = (const float*)d_in[1];   // (B,G,4)
    const int*   gt_classes = (const int*)  d_in[2];   // (B,G)
    const float* anchors    = (const float*)d_in[3];   // (N,2)
    float* out = (float*)d_out;

    // workspace carve-up (256B aligned)
    char* base = (char*)d_ws;
    size_t off = 0;
    auto carve = [&](size_t bytes) { char* p = base + off; off = (off + bytes + 255) & ~(size_t)255; return p; };
    unsigned long long* keys = (unsigned long long*)carve((size_t)BB * NN * 8);
    float* sp_sum            = (float*)carve((size_t)BB * NN * 4);
    unsigned char* cand      = (unsigned char*)carve((size_t)BB * NN);
    int* ncand               = (int*)carve(BB * 4);
    int* npos                = (int*)carve(BB * 4);
    float* box_part          = (float*)carve((size_t)BB * NB * 4);
    float* cls_part          = (float*)carve((size_t)BB * NB * 4);
    float* obj_part          = (float*)carve((size_t)BB * NB * 4);
    (void)in_sizes; (void)n_in; (void)out_size; (void)ws_size;

    k_init  <<<(BB * NN + 255) / 256, 256, 0, stream>>>(keys, ncand, npos);
    k_prep  <<<BB * NB,               256, 0, stream>>>(pred, sp_sum);
    k_cand  <<<BB * CBPI,             256, 0, stream>>>(anchors, gt_boxes, cand, ncand);
    k_assign<<<BB * GG,               256, 0, stream>>>(pred, gt_boxes, gt_classes,
                                                        sp_sum, cand, ncand, keys);
    k_losses<<<BB * NB,               256, 0, stream>>>(pred, gt_boxes, gt_classes, keys,
                                                        box_part, cls_part, obj_part, npos);
    k_final <<<1, 1, 0, stream>>>(box_part, cls_part, obj_part, npos, out);
}